// GraphAttention_15899968930183
// MI455X (gfx1250) — compile-verified
//
#include <hip/hip_runtime.h>

#define N_NODES 8192
#define F_IN    128
#define F_OUT   64
#define N_HEADS 4
#define NEG_INF (-1.0e10f)

typedef float v2f __attribute__((ext_vector_type(2)));
typedef float v8f __attribute__((ext_vector_type(8)));

static __device__ __forceinline__ v8f wmma_f32_16x16x4(v2f a, v2f b, v8f c) {
  // 8 args: (neg_a, A, neg_b, B, c_mod, C, reuse_a, reuse_b)
  return __builtin_amdgcn_wmma_f32_16x16x4_f32(false, a, false, b, (short)0, c,
                                               false, false);
}

static __device__ __forceinline__ float leaky02(float x) {
  return (x > 0.0f) ? x : 0.2f * x;
}

// CDNA5 async copy: copies 16 bytes from the global address in gaddr to the LDS address in ldsaddr, tracked by ASYNCcnt.
static __device__ __forceinline__ void async_load_b128(unsigned ldsaddr,
                                                       unsigned long long gaddr) {
  asm volatile("global_load_async_to_lds_b128 %0, %1, off"
               :: "v"(ldsaddr), "v"(gaddr) : "memory");
}

static __device__ __forceinline__ void wait_asynccnt0() {
  asm volatile("s_wait_asynccnt 0x0" ::: "memory");
}

// -------------------------------------------------------------------------
// K1: feats[h,n,o] = sum_k X[n,k] * W[h,k,o]    (WMMA f32 16x16xK4)
// One wave per 16x16 output tile. 8192 tiles total.
// -------------------------------------------------------------------------
__global__ void gat_feats_kernel(const float* __restrict__ X,
                                 const float* __restrict__ W,
                                 float* __restrict__ feats) {
  const int wave = (blockIdx.x * blockDim.x + threadIdx.x) >> 5;
  const int lane = threadIdx.x & 31;
  const int nt   = wave & 3;          // output col tile (F_OUT/16 = 4)
  const int rt   = (wave >> 2) & 511; // row tile (N/16 = 512)
  const int h    = wave >> 11;        // head
  const int half = lane >> 4;
  const int l16  = lane & 15;

  const int row  = rt * 16 + l16;     // A-operand row for this lane
  const int ocol = nt * 16 + l16;     // B-operand / output column
  const float* __restrict__ Wh = W + (size_t)h * (F_IN * F_OUT);

  v8f c = {};
#pragma unroll 4
  for (int kk = 0; kk < F_IN / 4; ++kk) {
    const int kbase = kk * 4 + half * 2;
    // A (16x4 f32): lane holds X[row, kbase], X[row, kbase+1]
    v2f a = *reinterpret_cast<const v2f*>(X + (size_t)row * F_IN + kbase);
    // B (4x16 f32): VGPR0 -> K=kbase, VGPR1 -> K=kbase+1, N=l16
    v2f b;
    b.x = Wh[(kbase + 0) * F_OUT + ocol];
    b.y = Wh[(kbase + 1) * F_OUT + ocol];
    c = wmma_f32_16x16x4(a, b, c);
  }

  float* __restrict__ fh = feats + (size_t)h * N_NODES * F_OUT;
#pragma unroll
  for (int j = 0; j < 8; ++j) {
    const int m = j + half * 8; // C/D layout: VGPR j -> row j (lanes<16) / j+8
    fh[(size_t)(rt * 16 + m) * F_OUT + ocol] = c[j];
  }
}

// -------------------------------------------------------------------------
// K2: e_self[h,n] = feats[h,n,:] . a_self[h,:]   (and e_neigh)
// One wave per (h,n). 32768 waves.
// -------------------------------------------------------------------------
__global__ void gat_evec_kernel(const float* __restrict__ feats,
                                const float* __restrict__ a_self,
                                const float* __restrict__ a_neigh,
                                float* __restrict__ e_self,
                                float* __restrict__ e_neigh) {
  const int wave = (blockIdx.x * blockDim.x + threadIdx.x) >> 5;
  const int lane = threadIdx.x & 31;
  const int h = wave / N_NODES;
  const int n = wave - h * N_NODES;

  const float* __restrict__ fr = feats + ((size_t)h * N_NODES + n) * F_OUT;
  const float f0 = fr[lane];
  const float f1 = fr[lane + 32];
  float s = f0 * a_self[h * F_OUT + lane] + f1 * a_self[h * F_OUT + lane + 32];
  float t = f0 * a_neigh[h * F_OUT + lane] + f1 * a_neigh[h * F_OUT + lane + 32];
#pragma unroll
  for (int off = 16; off >= 1; off >>= 1) {
    s += __shfl_xor(s, off, 32);
    t += __shfl_xor(t, off, 32);
  }
  if (lane == 0) {
    e_self[wave]  = s;
    e_neigh[wave] = t;
  }
}

// -------------------------------------------------------------------------
// K3: per-row online softmax stats: rowmax[h,n], rowsum[h,n]
// One wave per (h,n); lanes stride over the 8192 columns of A.
// -------------------------------------------------------------------------
__global__ void gat_rowstats_kernel(const float* __restrict__ A,
                                    const float* __restrict__ e_self,
                                    const float* __restrict__ e_neigh,
                                    float* __restrict__ rowmax,
                                    float* __restrict__ rowsum) {
  const int wave = (blockIdx.x * blockDim.x + threadIdx.x) >> 5;
  const int lane = threadIdx.x & 31;
  const int h = wave / N_NODES;
  const int n = wave - h * N_NODES;

  const float es = e_self[h * N_NODES + n];
  const float* __restrict__ Ar = A + (size_t)n * N_NODES;
  const float* __restrict__ en = e_neigh + (size_t)h * N_NODES;

  float mx = -3.0e38f, sm = 0.0f;
  for (int m = lane; m < N_NODES; m += 32) {
    const float a = Ar[m];
    float l = leaky02(es + en[m]);
    l += NEG_INF * (1.0f - a);
    const float nm = fmaxf(mx, l);
    sm = sm * __expf(mx - nm) + __expf(l - nm);
    mx = nm;
  }
#pragma unroll
  for (int off = 16; off >= 1; off >>= 1) {
    const float omx = __shfl_xor(mx, off, 32);
    const float osm = __shfl_xor(sm, off, 32);
    const float nm  = fmaxf(mx, omx);
    sm = sm * __expf(mx - nm) + osm * __expf(omx - nm);
    mx = nm;
  }
  if (lane == 0) {
    rowmax[wave] = mx;
    rowsum[wave] = sm;
  }
}

// -------------------------------------------------------------------------
// K4: out[n, h*64+o] = relu( (sum_m w[n,m] * feats[h,m,o]) / rowsum + b )
//     w[n,m] = exp(masked_leaky_logit - rowmax)   (0 for non-edges)
// Block = 8 waves, all same head, 8 adjacent 16-row tiles (128 rows).
// The 16x64 feats tile (4 KB) is async-copied to LDS once per block and
// double-buffered: prefetch of tile mt+1 overlaps WMMA on tile mt.
// -------------------------------------------------------------------------
__global__ void gat_attn_out_kernel(const float* __restrict__ A,
                                    const float* __restrict__ feats,
                                    const float* __restrict__ e_self,
                                    const float* __restrict__ e_neigh,
                                    const float* __restrict__ rowmax,
                                    const float* __restrict__ rowsum,
                                    const float* __restrict__ bias,
                                    float* __restrict__ out) {
  __shared__ float smem[2 * 16 * F_OUT];  // double-buffered 16x64 f32 tile

  const int lane = threadIdx.x & 31;
  const int wslot = threadIdx.x >> 5;              // wave in block: 0..7
  const int h    = blockIdx.x >> 6;                // head
  const int slab = blockIdx.x & 63;                // 128-row slab
  const int rt   = slab * 8 + wslot;               // this wave's 16-row tile
  const int half = lane >> 4;
  const int l16  = lane & 15;

  const int row  = rt * 16 + l16;
  const float es  = e_self[h * N_NODES + row];
  const float rmx = rowmax[h * N_NODES + row];
  const float* __restrict__ Ar = A + (size_t)row * N_NODES;
  const float* __restrict__ en = e_neigh + (size_t)h * N_NODES;
  const float* __restrict__ fh = feats + (size_t)h * N_NODES * F_OUT;

  // LDS byte offset of smem (low 32 bits of generic pointer == LDS offset)
  const unsigned lds_base = (unsigned)(uintptr_t)(&smem[0]);

  v8f acc0 = {}, acc1 = {}, acc2 = {}, acc3 = {};

  // Prologue: stage tile 0 into buffer 0 (each wave copies its 512B slice)
  {
    const unsigned long long g0 =
        (unsigned long long)(uintptr_t)(fh + wslot * 128 + lane * 4);
    async_load_b128(lds_base + (wslot * 128 + lane * 4) * 4, g0);
    wait_asynccnt0();
  }
  __syncthreads();

  int bufbase = 0;  // float index of active buffer in smem
  for (int mt = 0; mt < N_NODES / 16; ++mt) {
    const int mbase = mt * 16;
    const int nxtbase = bufbase ^ 1024;

    // Prefetch next feats tile into the other buffer (async, no wait yet)
    if (mt + 1 < N_NODES / 16) {
      const unsigned long long gnext = (unsigned long long)(uintptr_t)(
          fh + (size_t)(mbase + 16) * F_OUT + wslot * 128 + lane * 4);
      async_load_b128(lds_base + (nxtbase + wslot * 128 + lane * 4) * 4, gnext);
    }

    // Attention weights for this wave's 16 rows x 16 cols (A-operand tiles)
    v2f pw[4];
#pragma unroll
    for (int kk = 0; kk < 4; ++kk) {
      const int c0 = mbase + kk * 4 + half * 2;
      const v2f av = *reinterpret_cast<const v2f*>(Ar + c0);
      const v2f ev = *reinterpret_cast<const v2f*>(en + c0);
      float l0 = leaky02(es + ev.x) + NEG_INF * (1.0f - av.x);
      float l1 = leaky02(es + ev.y) + NEG_INF * (1.0f - av.y);
      v2f w;
      w.x = __expf(l0 - rmx);
      w.y = __expf(l1 - rmx);
      pw[kk] = w;
    }

    // 16 WMMAs: P(16x16) @ feats_tile(16x64), B operands from LDS
#pragma unroll
    for (int kk = 0; kk < 4; ++kk) {
      const int mloc = kk * 4 + half * 2;   // local row in the 16x64 LDS tile
      const int fb = bufbase + mloc * F_OUT + l16;
      v2f b0, b1, b2, b3;
      b0.x = smem[fb +  0];          b0.y = smem[fb + F_OUT +  0];
      b1.x = smem[fb + 16];          b1.y = smem[fb + F_OUT + 16];
      b2.x = smem[fb + 32];          b2.y = smem[fb + F_OUT + 32];
      b3.x = smem[fb + 48];          b3.y = smem[fb + F_OUT + 48];
      acc0 = wmma_f32_16x16x4(pw[kk], b0, acc0);
      acc1 = wmma_f32_16x16x4(pw[kk], b1, acc1);
      acc2 = wmma_f32_16x16x4(pw[kk], b2, acc2);
      acc3 = wmma_f32_16x16x4(pw[kk], b3, acc3);
    }

    // My slice of the next tile has landed; barrier makes all slices visible
    // and guarantees everyone is done reading the current buffer.
    wait_asynccnt0();
    __syncthreads();
    bufbase = nxtbase;
  }

  // Epilogue: normalize, add bias, ReLU, store to out[N, H*F_OUT]
  const float bv0 = bias[h * F_OUT + 0  + l16];
  const float bv1 = bias[h * F_OUT + 16 + l16];
  const float bv2 = bias[h * F_OUT + 32 + l16];
  const float bv3 = bias[h * F_OUT + 48 + l16];
#pragma unroll
  for (int j = 0; j < 8; ++j) {
    const int orow = rt * 16 + j + half * 8;
    const float inv = 1.0f / rowsum[h * N_NODES + orow];
    float* __restrict__ op = out + (size_t)orow * (N_HEADS * F_OUT) + h * F_OUT;
    op[0  + l16] = fmaxf(acc0[j] * inv + bv0, 0.0f);
    op[16 + l16] = fmaxf(acc1[j] * inv + bv1, 0.0f);
    op[32 + l16] = fmaxf(acc2[j] * inv + bv2, 0.0f);
    op[48 + l16] = fmaxf(acc3[j] * inv + bv3, 0.0f);
  }
}

// -------------------------------------------------------------------------
extern "C" void kernel_launch(void* const* d_in, const int* in_sizes, int n_in,
                              void* d_out, int out_size, void* d_ws, size_t ws_size,
                              hipStream_t stream) {
  (void)in_sizes; (void)n_in; (void)out_size; (void)ws_size;
  const float* X       = (const float*)d_in[0];
  const float* A       = (const float*)d_in[1];
  const float* W       = (const float*)d_in[2];
  const float* a_self  = (const float*)d_in[3];
  const float* a_neigh = (const float*)d_in[4];
  const float* bias    = (const float*)d_in[5];
  float* out = (float*)d_out;

  float* ws      = (float*)d_ws;
  float* feats   = ws;                                  // H*N*F_OUT = 2,097,152 f32
  float* e_self  = feats + (size_t)N_HEADS * N_NODES * F_OUT;
  float* e_neigh = e_self + (size_t)N_HEADS * N_NODES;  // 32768 each
  float* rmax    = e_neigh + (size_t)N_HEADS * N_NODES;
  float* rsum    = rmax + (size_t)N_HEADS * N_NODES;

  // K1: 8192 wave-tiles -> 1024 blocks x 256 threads (8 waves/block)
  hipLaunchKernelGGL(gat_feats_kernel, dim3(1024), dim3(256), 0, stream,
                     X, W, feats);
  // K2: 32768 waves -> 4096 blocks
  hipLaunchKernelGGL(gat_evec_kernel, dim3(4096), dim3(256), 0, stream,
                     feats, a_self, a_neigh, e_self, e_neigh);
  // K3: 32768 waves -> 4096 blocks
  hipLaunchKernelGGL(gat_rowstats_kernel, dim3(4096), dim3(256), 0, stream,
                     A, e_self, e_neigh, rmax, rsum);
  // K4: 256 blocks (4 heads x 64 slabs) x 256 threads (8 waves sharing LDS tile)
  hipLaunchKernelGGL(gat_attn_out_kernel, dim3(256), dim3(256), 0, stream,
                     A, feats, e_self, e_neigh, rmax, rsum, bias, out);
}